// ToyModel_44367012167682
// MI455X (gfx1250) — compile-verified
//
#include <hip/hip_runtime.h>

typedef __attribute__((ext_vector_type(2))) float v2f;
typedef __attribute__((ext_vector_type(4))) float v4f;
typedef __attribute__((ext_vector_type(8))) float v8f;

#define B_DIM   256
#define T_LEN   4096
#define D_DIM   64
#define U_DIM   2
#define NTHREADS 256
#define SPT     (T_LEN / NTHREADS)        // 16 timesteps per thread in scan phase
#define NWAVE   (NTHREADS / 32)           // 8 waves
#define TILES_PER_WAVE ((T_LEN / NWAVE) / 16)  // 32 tiles of 16 rows per wave

// XOR-swizzle an LDS float index: flips bank bits 1-4 with index bits 6-9.
// Kills the 128B-per-lane stride conflicts of the scan phase while keeping
// (u=0,u=1) pairs adjacent (bit 0 untouched) so b64 LDS accesses still work.
__device__ __forceinline__ int swz(int idx) {
  return idx ^ (((idx >> 6) & 15) << 1);
}

// Issue the 16 non-temporal b64 A-fragment loads for one 16-row tile.
__device__ __forceinline__ void load_frag(v2f (&frag)[16], const float* rowp) {
#pragma unroll
  for (int kc = 0; kc < 16; ++kc)
    frag[kc] = __builtin_nontemporal_load((const v2f*)(rowp + 4 * kc));
}

// 16 chained V_WMMA_F32_16X16X4_F32 (K=64) + swizzled LDS store of the
// U_DIM live output columns.
__device__ __forceinline__ void tile_mma_store(const v2f (&frag)[16],
                                               const v2f (&Bfrag)[16],
                                               float* hbuf, int t0, int n, int h) {
  v8f acc = {0.f, 0.f, 0.f, 0.f, 0.f, 0.f, 0.f, 0.f};
#pragma unroll
  for (int kc = 0; kc < 16; ++kc)
    acc = __builtin_amdgcn_wmma_f32_16x16x4_f32(
        /*neg_a=*/false, frag[kc], /*neg_b=*/false, Bfrag[kc],
        /*c_mod=*/(short)0, acc, /*reuse_a=*/false, /*reuse_b=*/false);
  // D layout: lane l, VGPR j holds D[M = j + 8*(l>>4), N = l&15].
  // Store address factorization: swz(2*(t0+j+8h)) + n == 2*t0 + ((16h+2j+n)^key)
  // because t0 is 16-aligned, the low offset is < 32, and key only has bits 1-4.
  const int key = ((t0 >> 5) & 15) << 1;   // uniform per tile
  const int lowbase = 16 * h + n;
  float* base = &hbuf[2 * t0];
  if (n < U_DIM) {
#pragma unroll
    for (int j = 0; j < 8; ++j)
      base[(lowbase + 2 * j) ^ key] = acc[j];
  }
}

__global__ __launch_bounds__(NTHREADS)
void rnn_fused_kernel(const float* __restrict__ x,
                      const float* __restrict__ W,
                      const float* __restrict__ Wr,
                      float* __restrict__ out)
{
  __shared__ float hbuf[T_LEN * U_DIM];   // 32 KB, swizzled h = x@W per batch row
  __shared__ float Ms[NTHREADS][5];       // 2x2 transfer matrices (padded stride)
  __shared__ float vs[NTHREADS][2];       // scan state vectors

  const int tid  = threadIdx.x;
  const int lane = tid & 31;
  const int wid  = tid >> 5;
  const int b    = blockIdx.x;
  const int n    = lane & 15;             // WMMA N index (column of D / B)
  const int h    = lane >> 4;             // half-wave select

  // ---------------- Phase 1: h = x[b] @ W via V_WMMA_F32_16X16X4_F32 -------
  // B fragment layout (4x16 f32): lanes 0-15 hold K=0,1 (VGPR0,1),
  // lanes 16-31 hold K=2,3. Pad columns n>=U with zeros.
  v2f Bfrag[16];
#pragma unroll
  for (int kc = 0; kc < 16; ++kc) {
    const int k0 = 4 * kc + 2 * h;
    float b0 = 0.0f, b1 = 0.0f;
    if (n < U_DIM) {
      b0 = W[k0 * U_DIM + n];
      b1 = W[(k0 + 1) * U_DIM + n];
    }
    Bfrag[kc].x = b0;
    Bfrag[kc].y = b1;
  }

  const float* xb = x + (size_t)b * T_LEN * D_DIM;
  // A fragment: lane holds row M = lane&15, K-pair 4*kc + 2*h + {0,1}
  const float* rowp0 = xb + (size_t)n * D_DIM + 2 * h;
  const int tw = wid * (T_LEN / NWAVE);   // this wave's first row

  // Double-buffered pipeline: prefetch the next tile's 16 fragments while the
  // current tile's WMMA chain executes (keeps 16-32 HBM loads in flight).
  v2f A0[16], A1[16];
  load_frag(A0, rowp0 + (size_t)tw * D_DIM);
  for (int tp = 0; tp < TILES_PER_WAVE - 2; tp += 2) {
    const int t0 = tw + tp * 16;
    load_frag(A1, rowp0 + (size_t)(t0 + 16) * D_DIM);
    tile_mma_store(A0, Bfrag, hbuf, t0, n, h);
    load_frag(A0, rowp0 + (size_t)(t0 + 32) * D_DIM);
    tile_mma_store(A1, Bfrag, hbuf, t0 + 16, n, h);
  }
  {
    const int t0 = tw + (TILES_PER_WAVE - 2) * 16;
    load_frag(A1, rowp0 + (size_t)(t0 + 16) * D_DIM);
    tile_mma_store(A0, Bfrag, hbuf, t0, n, h);
    tile_mma_store(A1, Bfrag, hbuf, t0 + 16, n, h);
  }
  __syncthreads();

  // ---------------- Phase 2: exact parallel linear scan ---------------------
  // out_t = h_t + out_{t-1} @ Wr  (row-vector convention). Per-thread chunk of
  // SPT steps is the affine map e -> e @ Wr^SPT + v; Hillis-Steele combines
  // the 256 maps, then a seeded re-scan produces exact outputs.
  const float A00 = Wr[0], A01 = Wr[1], A10 = Wr[2], A11 = Wr[3];
  const int tbase = tid * SPT;

  // Local scan with zero entering state.
  float p0 = 0.0f, p1 = 0.0f;
#pragma unroll
  for (int k = 0; k < SPT; ++k) {
    const int t = tbase + k;
    v2f hv = *(const v2f*)&hbuf[swz(t * U_DIM)];
    const float o0 = hv.x + p0 * A00 + p1 * A10;
    const float o1 = hv.y + p0 * A01 + p1 * A11;
    p0 = o0; p1 = o1;
  }

  // Wr^SPT (= Wr^16) by repeated squaring.
  float M00 = A00, M01 = A01, M10 = A10, M11 = A11;
#pragma unroll
  for (int s = 0; s < 4; ++s) {
    const float t00 = M00 * M00 + M01 * M10;
    const float t01 = M00 * M01 + M01 * M11;
    const float t10 = M10 * M00 + M11 * M10;
    const float t11 = M10 * M01 + M11 * M11;
    M00 = t00; M01 = t01; M10 = t10; M11 = t11;
  }

  // Inclusive Hillis-Steele scan of (M, v): combined(e) = e@(Ml Mr) + (vl@Mr + vr)
  float c00 = M00, c01 = M01, c10 = M10, c11 = M11;
  float cv0 = p0, cv1 = p1;
  Ms[tid][0] = c00; Ms[tid][1] = c01; Ms[tid][2] = c10; Ms[tid][3] = c11;
  vs[tid][0] = cv0; vs[tid][1] = cv1;
  __syncthreads();
  for (int d = 1; d < NTHREADS; d <<= 1) {
    float l00 = 0.f, l01 = 0.f, l10 = 0.f, l11 = 0.f, lv0 = 0.f, lv1 = 0.f;
    const bool has = (tid >= d);
    if (has) {
      l00 = Ms[tid - d][0]; l01 = Ms[tid - d][1];
      l10 = Ms[tid - d][2]; l11 = Ms[tid - d][3];
      lv0 = vs[tid - d][0]; lv1 = vs[tid - d][1];
    }
    __syncthreads();
    if (has) {
      const float nv0 = lv0 * c00 + lv1 * c10 + cv0;
      const float nv1 = lv0 * c01 + lv1 * c11 + cv1;
      const float n00 = l00 * c00 + l01 * c10;
      const float n01 = l00 * c01 + l01 * c11;
      const float n10 = l10 * c00 + l11 * c10;
      const float n11 = l10 * c01 + l11 * c11;
      c00 = n00; c01 = n01; c10 = n10; c11 = n11;
      cv0 = nv0; cv1 = nv1;
      Ms[tid][0] = c00; Ms[tid][1] = c01; Ms[tid][2] = c10; Ms[tid][3] = c11;
      vs[tid][0] = cv0; vs[tid][1] = cv1;
    }
    __syncthreads();
  }

  // Exclusive value = state entering this thread's chunk.
  float e0 = 0.0f, e1 = 0.0f;
  if (tid > 0) { e0 = vs[tid - 1][0]; e1 = vs[tid - 1][1]; }

  // Seeded re-scan; write final outputs directly (contiguous 128B per thread).
  float* outp = out + (size_t)b * T_LEN * U_DIM;
  float p0b = e0, p1b = e1;
#pragma unroll
  for (int k2 = 0; k2 < SPT / 2; ++k2) {
    const int t = tbase + 2 * k2;
    v2f ha = *(const v2f*)&hbuf[swz(t * U_DIM)];
    const float o0 = ha.x + p0b * A00 + p1b * A10;
    const float o1 = ha.y + p0b * A01 + p1b * A11;
    v2f hbv = *(const v2f*)&hbuf[swz((t + 1) * U_DIM)];
    const float q0 = hbv.x + o0 * A00 + o1 * A10;
    const float q1 = hbv.y + o0 * A01 + o1 * A11;
    v4f ov; ov.x = o0; ov.y = o1; ov.z = q0; ov.w = q1;
    __builtin_nontemporal_store(ov, (v4f*)(outp + t * U_DIM));
    p0b = q0; p1b = q1;
  }
}

extern "C" void kernel_launch(void* const* d_in, const int* in_sizes, int n_in,
                              void* d_out, int out_size, void* d_ws, size_t ws_size,
                              hipStream_t stream) {
  (void)in_sizes; (void)n_in; (void)out_size; (void)d_ws; (void)ws_size;
  const float* x  = (const float*)d_in[0];
  const float* W  = (const float*)d_in[1];
  const float* Wr = (const float*)d_in[2];
  float* out = (float*)d_out;
  rnn_fused_kernel<<<dim3(B_DIM), dim3(NTHREADS), 0, stream>>>(x, W, Wr, out);
}